// GATNet_69191923138731
// MI455X (gfx1250) — compile-verified
//
#include <hip/hip_runtime.h>
#include <hip/hip_bf16.h>
#include <stdint.h>

#define N_LEFT   20000
#define N_RIGHT  20000
#define N_EDGE   300000
#define CDIM     256
#define HC       512      // H*C
#define NEG_SLOPE 0.2f

typedef __attribute__((ext_vector_type(2))) float v2f;
typedef __attribute__((ext_vector_type(8))) float v8f;

// ---------------- workspace layout (float offsets) ----------------
#define OFF_UDST   0                                   // 512
#define OFF_UEDGE  512                                 // 512
#define OFF_ASRC   1024                                // 40000
#define OFF_ADST   (OFF_ASRC + 2*N_LEFT)               // 41024
#define OFF_MENC   (OFF_ADST + 2*N_RIGHT)              // 81024 (as unsigned)
#define OFF_DEN    (OFF_MENC + 2*N_RIGHT)              // 121024
#define OFF_ALPHA  (OFF_DEN  + 2*N_RIGHT)              // 161024
#define OFF_XSRC   (OFF_ALPHA + 2*(size_t)N_EDGE)      // 761024
#define OFF_AGG    (OFF_XSRC + (size_t)N_LEFT*HC)      // 11001024
#define OFF_H1     (OFF_AGG  + (size_t)N_RIGHT*HC)     // 21241024
#define OFF_H2     (OFF_H1   + (size_t)N_RIGHT*CDIM)   // 26361024

#define ENC_NEG_INF 0x007FFFFFu

__device__ inline unsigned fenc(float f) {
  unsigned u = __float_as_uint(f);
  return (u & 0x80000000u) ? ~u : (u | 0x80000000u);
}
__device__ inline float fdec(unsigned u) {
  return (u & 0x80000000u) ? __uint_as_float(u ^ 0x80000000u)
                           : __uint_as_float(~u);
}
__device__ inline float wave_sum(float v) {
  #pragma unroll
  for (int m = 16; m >= 1; m >>= 1) v += __shfl_xor(v, m, 32);
  return v;
}

// ---------------- init: zero agg, init segment-max/den ----------------
__global__ __launch_bounds__(256) void init_kernel(float* agg, unsigned* m_enc,
                                                   float* den) {
  int t = blockIdx.x * 256 + threadIdx.x;
  if (t < N_RIGHT * HC) agg[t] = 0.f;
  if (t < 2 * N_RIGHT) { m_enc[t] = ENC_NEG_INF; den[t] = 0.f; }
}

// ---------------- fold att_dst into W, att_edge into W_edge ----------------
__global__ __launch_bounds__(512) void fold_kernel(
    const float* __restrict__ W, const float* __restrict__ att_dst,
    const float* __restrict__ W_edge, const float* __restrict__ att_edge,
    float* __restrict__ u_dst, float* __restrict__ u_edge) {
  int t = threadIdx.x;         // 512 threads: (k, h)
  int k = t >> 1, h = t & 1;
  const float* wrow  = W      + (size_t)k * HC + h * CDIM;
  const float* werow = W_edge + (size_t)k * HC + h * CDIM;
  const float* ad = att_dst  + h * CDIM;
  const float* ae = att_edge + h * CDIM;
  float sd = 0.f, se = 0.f;
  for (int c = 0; c < CDIM; ++c) { sd += wrow[c] * ad[c]; se += werow[c] * ae[c]; }
  u_dst[2 * k + h] = sd;
  u_edge[2 * k + h] = se;
}

// ---------------- fp32 WMMA GEMM: Out[M,N] = f(A)[M,K] @ B[K,N] (+biasO) ---
// MODE 0: A plain                     (x_src = left @ W)
// MODE 1: A = relu(A0 + biasA[k])     (h1 = relu(agg+bias_conv) @ W_lt + b)
// MODE 2: A = relu(A0)                (h2 = relu(h1) @ W_fm + b)
// MODE 3: A = concat(A0[*,0:256], A1) (upd = [right|h2] @ W_pc + b)
template <int MODE, int N, int K>
__global__ __launch_bounds__(256) void gemm_wmma(
    const float* __restrict__ A0, const float* __restrict__ A1,
    const float* __restrict__ Bm, const float* __restrict__ biasA,
    const float* __restrict__ biasO, float* __restrict__ Out, int M) {
  constexpr int NTN = N / 64;                    // 64-wide N tiles
  const int lane = threadIdx.x & 31;
  const int gw   = blockIdx.x * 8 + (threadIdx.x >> 5);
  const int tm   = gw / NTN;
  const int tn   = gw % NTN;
  if (tm * 16 >= M) return;                      // wave-uniform: EXEC stays full

  const int row  = tm * 16 + (lane & 15);
  const int koff = (lane >> 4) << 1;             // 0 or 2
  const int ncol = tn * 64 + (lane & 15);

  v8f acc0 = {}, acc1 = {}, acc2 = {}, acc3 = {};

  const float* bptr = Bm + (size_t)koff * N + ncol;

  auto step = [&](v2f a, const float* __restrict__ bp) {
    v2f b0, b1, b2, b3;
    b0.x = bp[0];      b0.y = bp[N];
    b1.x = bp[16];     b1.y = bp[N + 16];
    b2.x = bp[32];     b2.y = bp[N + 32];
    b3.x = bp[48];     b3.y = bp[N + 48];
    acc0 = __builtin_amdgcn_wmma_f32_16x16x4_f32(false, a, false, b0, (short)0, acc0, false, false);
    acc1 = __builtin_amdgcn_wmma_f32_16x16x4_f32(false, a, false, b1, (short)0, acc1, false, false);
    acc2 = __builtin_amdgcn_wmma_f32_16x16x4_f32(false, a, false, b2, (short)0, acc2, false, false);
    acc3 = __builtin_amdgcn_wmma_f32_16x16x4_f32(false, a, false, b3, (short)0, acc3, false, false);
  };

  if (MODE == 3) {
    // first half of K: right_features, second half: h2
    const float* ap = A0 + (size_t)row * CDIM + koff;
    #pragma unroll 4
    for (int k0 = 0; k0 < CDIM; k0 += 4) {
      step(*(const v2f*)ap, bptr);
      ap += 4; bptr += 4 * N;
    }
    ap = A1 + (size_t)row * CDIM + koff;
    #pragma unroll 4
    for (int k0 = 0; k0 < CDIM; k0 += 4) {
      step(*(const v2f*)ap, bptr);
      ap += 4; bptr += 4 * N;
    }
  } else {
    const float* ap = A0 + (size_t)row * K + koff;
    const float* bA = (MODE == 1) ? (biasA + koff) : nullptr;
    #pragma unroll 4
    for (int k0 = 0; k0 < K; k0 += 4) {
      v2f t = *(const v2f*)ap;
      v2f a;
      if (MODE == 1) {
        float x0 = t.x + bA[0], x1 = t.y + bA[1];
        a.x = x0 > 0.f ? x0 : 0.f;  a.y = x1 > 0.f ? x1 : 0.f;
        bA += 4;
      } else if (MODE == 2) {
        a.x = t.x > 0.f ? t.x : 0.f;  a.y = t.y > 0.f ? t.y : 0.f;
      } else {
        a = t;
      }
      step(a, bptr);
      ap += 4; bptr += 4 * N;
    }
  }

  float bo0 = 0.f, bo1 = 0.f, bo2 = 0.f, bo3 = 0.f;
  if (biasO) { bo0 = biasO[ncol]; bo1 = biasO[ncol + 16];
               bo2 = biasO[ncol + 32]; bo3 = biasO[ncol + 48]; }
  const int rbase = tm * 16 + ((lane >> 4) << 3);   // lanes 16-31 hold M=8..15
  #pragma unroll
  for (int r = 0; r < 8; ++r) {
    size_t o = (size_t)(rbase + r) * N + ncol;
    Out[o]      = acc0[r] + bo0;
    Out[o + 16] = acc1[r] + bo1;
    Out[o + 32] = acc2[r] + bo2;
    Out[o + 48] = acc3[r] + bo3;
  }
}

// ---------------- per-node attention logits (a_src from x_src, a_dst folded)
__global__ __launch_bounds__(256) void node_logits_kernel(
    const float* __restrict__ x_src, const float* __restrict__ right,
    const float* __restrict__ att_src, const float* __restrict__ u_dst,
    float* __restrict__ a_src, float* __restrict__ a_dst) {
  const int lane = threadIdx.x & 31;
  const int n = blockIdx.x * 8 + (threadIdx.x >> 5);
  const float* xs = x_src + (size_t)n * HC;
  const float* rf = right + (size_t)n * CDIM;
  float ss0 = 0.f, ss1 = 0.f, sd0 = 0.f, sd1 = 0.f;
  #pragma unroll
  for (int i = 0; i < 8; ++i) {
    int c = lane + 32 * i;
    ss0 += xs[c] * att_src[c];
    ss1 += xs[CDIM + c] * att_src[CDIM + c];
    float r = rf[c];
    sd0 += r * u_dst[2 * c];
    sd1 += r * u_dst[2 * c + 1];
  }
  ss0 = wave_sum(ss0); ss1 = wave_sum(ss1);
  sd0 = wave_sum(sd0); sd1 = wave_sum(sd1);
  if (lane == 0) {
    a_src[2 * n] = ss0; a_src[2 * n + 1] = ss1;
    a_dst[2 * n] = sd0; a_dst[2 * n + 1] = sd1;
  }
}

// ---------------- per-edge logits + leaky relu + segment max --------------
__global__ __launch_bounds__(256) void edge_logits_kernel(
    const float* __restrict__ edge_feat, const int* __restrict__ eidx,
    const float* __restrict__ u_edge, const float* __restrict__ a_src,
    const float* __restrict__ a_dst, float* __restrict__ alpha_ws,
    unsigned* __restrict__ m_enc) {
  __shared__ float s_u[2 * CDIM];
  for (int i = threadIdx.x; i < 2 * CDIM; i += 256) s_u[i] = u_edge[i];
  __syncthreads();
  const int lane = threadIdx.x & 31;
  const int e = blockIdx.x * 8 + (threadIdx.x >> 5);
  const float* ef = edge_feat + (size_t)e * CDIM;
  float s0 = 0.f, s1 = 0.f;
  #pragma unroll
  for (int i = 0; i < 8; ++i) {
    int k = lane + 32 * i;
    float v = ef[k];
    s0 += v * s_u[2 * k];
    s1 += v * s_u[2 * k + 1];
  }
  s0 = wave_sum(s0); s1 = wave_sum(s1);
  if (lane == 0) {
    int src = eidx[e], dst = eidx[N_EDGE + e];
    float al0 = a_src[2 * src] + a_dst[2 * dst] + s0;
    float al1 = a_src[2 * src + 1] + a_dst[2 * dst + 1] + s1;
    al0 = al0 >= 0.f ? al0 : NEG_SLOPE * al0;
    al1 = al1 >= 0.f ? al1 : NEG_SLOPE * al1;
    alpha_ws[2 * e] = al0; alpha_ws[2 * e + 1] = al1;
    atomicMax(&m_enc[2 * dst], fenc(al0));
    atomicMax(&m_enc[2 * dst + 1], fenc(al1));
  }
}

// ---------------- exp(alpha - max) and segment denominator ----------------
__global__ __launch_bounds__(256) void expden_kernel(
    const int* __restrict__ eidx, float* __restrict__ alpha_ws,
    const unsigned* __restrict__ m_enc, float* __restrict__ den) {
  int t = blockIdx.x * 256 + threadIdx.x;
  if (t >= 2 * N_EDGE) return;
  int e = t >> 1, h = t & 1;
  int dst = eidx[N_EDGE + e];
  float ex = expf(alpha_ws[t] - fdec(m_enc[2 * dst + h]));
  alpha_ws[t] = ex;
  atomicAdd(&den[2 * dst + h], ex);
}

// ---------------- normalize alpha, emit alpha output, scatter messages ----
__global__ __launch_bounds__(256) void aggregate_kernel(
    const int* __restrict__ eidx, const float* __restrict__ alpha_ws,
    const float* __restrict__ den, const float* __restrict__ x_src,
    float* __restrict__ agg, float* __restrict__ out_alpha) {
  const int lane = threadIdx.x & 31;
  const int e = blockIdx.x * 8 + (threadIdx.x >> 5);
  const int src = eidx[e], dst = eidx[N_EDGE + e];
  const float a0 = alpha_ws[2 * e]     / (den[2 * dst]     + 1e-16f);
  const float a1 = alpha_ws[2 * e + 1] / (den[2 * dst + 1] + 1e-16f);
  if (lane == 0) { out_alpha[2 * e] = a0; out_alpha[2 * e + 1] = a1; }
  const float* xs = x_src + (size_t)src * HC;
  float* og = agg + (size_t)dst * HC;
  #pragma unroll
  for (int i = 0; i < 16; ++i) {
    int idx = lane + 32 * i;                 // 0..511, coalesced
    float a = (idx < CDIM) ? a0 : a1;
    atomicAdd(&og[idx], xs[idx] * a);
  }
}

extern "C" void kernel_launch(void* const* d_in, const int* in_sizes, int n_in,
                              void* d_out, int out_size, void* d_ws, size_t ws_size,
                              hipStream_t stream) {
  const float* left      = (const float*)d_in[0];
  const float* right     = (const float*)d_in[1];
  const float* edgef     = (const float*)d_in[2];
  const int*   eidx      = (const int*)d_in[3];
  const float* W         = (const float*)d_in[4];
  const float* W_edge    = (const float*)d_in[5];
  const float* att_src   = (const float*)d_in[6];
  const float* att_dst   = (const float*)d_in[7];
  const float* att_edge  = (const float*)d_in[8];
  const float* bias_conv = (const float*)d_in[9];
  const float* W_lt      = (const float*)d_in[10];
  const float* b_lt      = (const float*)d_in[11];
  const float* W_fm      = (const float*)d_in[12];
  const float* b_fm      = (const float*)d_in[13];
  const float* W_pc      = (const float*)d_in[14];
  const float* b_pc      = (const float*)d_in[15];

  float*    ws       = (float*)d_ws;
  float*    u_dst    = ws + OFF_UDST;
  float*    u_edge   = ws + OFF_UEDGE;
  float*    a_src    = ws + OFF_ASRC;
  float*    a_dst    = ws + OFF_ADST;
  unsigned* m_enc    = (unsigned*)(ws + OFF_MENC);
  float*    den      = ws + OFF_DEN;
  float*    alpha_ws = ws + OFF_ALPHA;
  float*    x_src    = ws + OFF_XSRC;
  float*    agg      = ws + OFF_AGG;
  float*    h1       = ws + OFF_H1;
  float*    h2       = ws + OFF_H2;

  float* out_upd   = (float*)d_out;
  float* out_alpha = out_upd + (size_t)N_RIGHT * CDIM;

  // 1) init segment-softmax state + aggregation buffer
  init_kernel<<<(N_RIGHT * HC + 255) / 256, 256, 0, stream>>>(agg, m_enc, den);
  // 2) fold attention vectors through the weight matrices
  fold_kernel<<<1, 512, 0, stream>>>(W, att_dst, W_edge, att_edge, u_dst, u_edge);
  // 3) x_src = left @ W  (WMMA f32, 20000x512x256)
  gemm_wmma<0, HC, CDIM><<<1250, 256, 0, stream>>>(left, nullptr, W, nullptr,
                                                   nullptr, x_src, N_LEFT);
  // 4) per-node logits
  node_logits_kernel<<<N_LEFT / 8, 256, 0, stream>>>(x_src, right, att_src,
                                                     u_dst, a_src, a_dst);
  // 5) per-edge logits + leaky relu + segment max
  edge_logits_kernel<<<N_EDGE / 8, 256, 0, stream>>>(edgef, eidx, u_edge,
                                                     a_src, a_dst, alpha_ws, m_enc);
  // 6) exp + segment sum
  expden_kernel<<<(2 * N_EDGE + 255) / 256, 256, 0, stream>>>(eidx, alpha_ws,
                                                              m_enc, den);
  // 7) normalize + alpha output + message scatter-add (L2-resident)
  aggregate_kernel<<<N_EDGE / 8, 256, 0, stream>>>(eidx, alpha_ws, den, x_src,
                                                   agg, out_alpha);
  // 8) h1 = relu(agg + bias_conv) @ W_lt + b_lt
  gemm_wmma<1, CDIM, HC><<<625, 256, 0, stream>>>(agg, nullptr, W_lt, bias_conv,
                                                  b_lt, h1, N_RIGHT);
  // 9) h2 = relu(h1) @ W_fm + b_fm
  gemm_wmma<2, CDIM, CDIM><<<625, 256, 0, stream>>>(h1, nullptr, W_fm, nullptr,
                                                    b_fm, h2, N_RIGHT);
  // 10) upd = [right | h2] @ W_pc + b_pc -> d_out
  gemm_wmma<3, CDIM, HC><<<625, 256, 0, stream>>>(right, h2, W_pc, nullptr,
                                                  b_pc, out_upd, N_RIGHT);
}